// SimulateConv1d_38079180046569
// MI455X (gfx1250) — compile-verified
//
#include <hip/hip_runtime.h>

typedef float v2f __attribute__((ext_vector_type(2)));
typedef float v8f __attribute__((ext_vector_type(8)));
typedef int   v4i __attribute__((vector_size(16)));   // matches builtin param type

static constexpr int LEN  = 2048;   // sequence length (== out_size per row)
static constexpr int ROWS = 512;    // B*H = 8*64
static constexpr int HID  = 128;
static constexpr int KSZ  = 9;
static constexpr int OT_PER_WAVE   = 16;  // 16 o-tiles * 16 = 256 outputs / wave
static constexpr int WAVES_PER_BLK = 8;
static constexpr int WBUF = 272;    // 4 halo + 256 strip + 4 halo + pad (16B mult)

__global__ __launch_bounds__(256)
void conv1d_mlp_wmma(const float* __restrict__ x,
                     const float* __restrict__ W1,
                     const float* __restrict__ b1,
                     const float* __restrict__ W2,
                     const float* __restrict__ b2,
                     float* __restrict__ out) {
  __shared__ float wbuf[WAVES_PER_BLK][WBUF];

  const int tid    = threadIdx.x;
  const int lane   = tid & 31;
  const int wave   = tid >> 5;
  const int waveId = blockIdx.x * WAVES_PER_BLK + wave;
  const int row    = waveId >> 3;      // 0..511
  const int strip  = waveId & 7;       // 256-wide strip within the row

  const float* __restrict__ xr   = x   + row * LEN;
  float* __restrict__       outr = out + row * LEN;

  const int m    = lane & 15;          // A: M index / B,D: N index
  const int hi   = lane >> 4;          // half-wave select
  const int koff = hi << 1;            // fp32 WMMA: upper half holds K+2,K+3

  // ------------------------------------------------------------------
  // Stage the strip window into LDS: wbuf[wave][j] = x[strip*256 + j - 4]
  // (zero outside the row). Valid j span is [jlo, jhi), multiples of 4.
  // ------------------------------------------------------------------
  const int jlo = (strip == 0) ? 4 : 0;
  const int jhi = (strip == 7) ? 260 : 264;

#if __has_builtin(__builtin_amdgcn_global_load_async_to_lds_b128)
  // zero the halo / tail slots (DS ops stay in-order within the wave)
#pragma unroll
  for (int i = 0; i < 9; ++i) {
    int j = lane + i * 32;
    if (j < WBUF && (j < jlo || j >= jhi)) wbuf[wave][j] = 0.0f;
  }
  // async-copy the valid span, 16B per lane per issue
  {
    const int nchunks = (jhi - jlo) >> 2;          // 65 or 66
    const float* gsrc = xr + (strip * 256 - 4);    // x index of j == 0
#pragma unroll
    for (int i = 0; i < 3; ++i) {
      int chunk = lane + i * 32;
      if (chunk < nchunks) {
        int j = jlo + chunk * 4;
        __builtin_amdgcn_global_load_async_to_lds_b128(
            (__attribute__((address_space(1))) v4i*)(gsrc + j),
            (__attribute__((address_space(3))) v4i*)&wbuf[wave][j],
            0, 0);
      }
    }
  }
#if __has_builtin(__builtin_amdgcn_s_wait_asynccnt)
  __builtin_amdgcn_s_wait_asynccnt(0);
#else
  asm volatile("s_wait_asynccnt 0" ::: "memory");
#endif
  asm volatile("" ::: "memory");   // keep LDS reads below the wait
#else
  // fallback: load + ds_store
#pragma unroll
  for (int i = 0; i < 9; ++i) {
    int j = lane + i * 32;
    if (j < WBUF) {
      float v = 0.0f;
      if (j >= jlo && j < jhi) v = xr[strip * 256 + j - 4];
      wbuf[wave][j] = v;
    }
  }
#endif

  // ---- W1^T (+ b1 in the K=9 slot) in fp32-WMMA A-layout, kept in VGPRs ----
  v2f a[8][3];
#pragma unroll
  for (int t = 0; t < 8; ++t) {
#pragma unroll
    for (int s = 0; s < 3; ++s) {
      const int d  = t * 16 + m;
      const int k0 = 4 * s + koff;
      const int k1 = k0 + 1;
      a[t][s].x = (k0 < KSZ) ? W1[k0 * HID + d] : 0.0f;
      a[t][s].y = (k1 < KSZ) ? W1[k1 * HID + d]
                             : ((k1 == KSZ) ? b1[d] : 0.0f);  // bias slot K=9
    }
  }

  // ---- W2 in the D-matrix per-lane layout (VGPR r <-> hidden t*16+r+8*hi) ----
  float w2v[8][8];
#pragma unroll
  for (int t = 0; t < 8; ++t)
#pragma unroll
    for (int r = 0; r < 8; ++r)
      w2v[t][r] = W2[t * 16 + r + hi * 8];

  const float bias2  = b2[0];
  const float oneSel = hi ? 0.0f : 1.0f;   // B value for the K=9 bias column
  const float inf    = __builtin_inff();
  const float* wl    = &wbuf[wave][0];

  for (int ot = 0; ot < OT_PER_WAVE; ++ot) {
    const int o0   = strip * 256 + ot * 16;
    const int base = ot * 16 + m + koff;   // LDS index for k = koff

    // B-matrix tile from LDS: unconditional, halo already zero-padded
    v2f bm0 = { wl[base],     wl[base + 1] };
    v2f bm1 = { wl[base + 4], wl[base + 5] };
    v2f bm2 = { wl[base + 8], oneSel };    // K=8 tap + bias-ones column

    float acc0 = 0.0f, acc1 = 0.0f, acc2 = 0.0f, acc3 = 0.0f;
#pragma unroll
    for (int t = 0; t < 8; ++t) {
      v8f c = {};
      c = __builtin_amdgcn_wmma_f32_16x16x4_f32(false, a[t][0], false, bm0,
                                                (short)0, c, false, false);
      c = __builtin_amdgcn_wmma_f32_16x16x4_f32(false, a[t][1], false, bm1,
                                                (short)0, c, false, false);
      c = __builtin_amdgcn_wmma_f32_16x16x4_f32(false, a[t][2], false, bm2,
                                                (short)0, c, false, false);
      // relu + dot(W2): v_med3_f32 is a 1-op ReLU; 4 accumulators break the chain
      acc0 = fmaf(__builtin_amdgcn_fmed3f(0.0f, c[0], inf), w2v[t][0], acc0);
      acc1 = fmaf(__builtin_amdgcn_fmed3f(0.0f, c[1], inf), w2v[t][1], acc1);
      acc2 = fmaf(__builtin_amdgcn_fmed3f(0.0f, c[2], inf), w2v[t][2], acc2);
      acc3 = fmaf(__builtin_amdgcn_fmed3f(0.0f, c[3], inf), w2v[t][3], acc3);
      acc0 = fmaf(__builtin_amdgcn_fmed3f(0.0f, c[4], inf), w2v[t][4], acc0);
      acc1 = fmaf(__builtin_amdgcn_fmed3f(0.0f, c[5], inf), w2v[t][5], acc1);
      acc2 = fmaf(__builtin_amdgcn_fmed3f(0.0f, c[6], inf), w2v[t][6], acc2);
      acc3 = fmaf(__builtin_amdgcn_fmed3f(0.0f, c[7], inf), w2v[t][7], acc3);
    }

    float acc = (acc0 + acc1) + (acc2 + acc3);
    // lanes l and l+16 hold complementary hidden halves of the same output
    acc += __shfl_xor(acc, 16, 32);
    if (hi == 0) outr[o0 + m] = acc + bias2;
  }
}

extern "C" void kernel_launch(void* const* d_in, const int* in_sizes, int n_in,
                              void* d_out, int out_size, void* d_ws, size_t ws_size,
                              hipStream_t stream) {
  const float* x  = (const float*)d_in[0];
  const float* W1 = (const float*)d_in[1];
  const float* b1 = (const float*)d_in[2];
  const float* W2 = (const float*)d_in[3];
  const float* b2 = (const float*)d_in[4];
  float* out = (float*)d_out;

  const int total_waves = ROWS * 8;                 // 4096
  dim3 grid(total_waves / WAVES_PER_BLK);           // 512 blocks
  dim3 block(32 * WAVES_PER_BLK);                   // 256 threads = 8 waves
  conv1d_mlp_wmma<<<grid, block, 0, stream>>>(x, W1, b1, W2, b2, out);
}